// TransitionUp_40613210751457
// MI455X (gfx1250) — compile-verified
//
#include <hip/hip_runtime.h>

// ---------------------------------------------------------------------------
// CDNA5 (gfx1250) implementation of TransitionUp:
//   down_f  = down_features @ W_down.T + b_down        (WMMA f32 16x16x4)
//   up_proj = up_features   @ W_up.T   + b_up          (WMMA f32 16x16x4)
//   out     = up_proj + IDW-interp(3-NN of up_points in down_points, down_f)
// ---------------------------------------------------------------------------

typedef __attribute__((ext_vector_type(2))) float v2f;
typedef __attribute__((ext_vector_type(8))) float v8f;

// One wave32 computes a 16x(NT*16) slab of  out[M,128] = in[M,KD] @ W^T + b,
// with W stored [128, KD] row-major (torch nn.Linear layout), so
// B[k][n] = W[n][k]: both A and B are contiguous float2 row reads.
//
// NT accumulators => NT independent WMMA dependency chains per wave, and the
// A fragment is loaded once per k-step instead of once per column tile.
//
// V_WMMA_F32_16X16X4_F32 operand layout (wave32):
//   A 16x4 : lane l -> row (l&15), VGPR{0,1} = K = 2*(l>>4) + {0,1}
//   B 4x16 : lane l -> col (l&15), VGPR{0,1} = K = 2*(l>>4) + {0,1}
//   C/D    : VGPR r, lane l -> row r + 8*(l>>4), col (l&15)
template <int KD, int NT>
__global__ __launch_bounds__(128) void gemm_bias_wmma_f32(
    const float* __restrict__ in,    // [Mrows, KD]
    const float* __restrict__ W,     // [128, KD]
    const float* __restrict__ bias,  // [128]
    float* __restrict__ out,         // [Mrows, 128]
    int Mrows) {
  constexpr int COL_TILES = 128 / (NT * 16);
  const int lane = threadIdx.x & 31;
  const int wave = threadIdx.x >> 5;
  const int slab = blockIdx.x * 4 + wave;   // 4 waves per block, 1 slab each
  const int tm   = slab / COL_TILES;
  const int tc   = slab % COL_TILES;
  const int m0   = tm * 16;
  if (m0 >= Mrows) return;                  // wave-uniform: EXEC stays all-1s
  const int n0   = tc * (NT * 16);

  const int l15  = lane & 15;
  const int half = lane >> 4;               // 0: K=0,1   1: K=2,3
  const int koff = half * 2;

  const float* arow = in + (size_t)(m0 + l15) * KD + koff;
  const float* brow = W  + (size_t)(n0 + l15) * KD + koff;

  v8f acc[NT];
#pragma unroll
  for (int t = 0; t < NT; ++t) acc[t] = (v8f){};

#pragma unroll 2
  for (int k = 0; k < KD; k += 4) {
    const v2f a = *(const v2f*)(arow + k);
#pragma unroll
    for (int t = 0; t < NT; ++t) {
      const v2f b = *(const v2f*)(brow + (size_t)t * 16 * KD + k);
      // 8 args: (neg_a, A, neg_b, B, c_mod, C, reuse_a, reuse_b)
      acc[t] = __builtin_amdgcn_wmma_f32_16x16x4_f32(
          false, a, false, b, (short)0, acc[t], false, false);
    }
  }

  float* orow = out + (size_t)(m0 + half * 8) * 128 + l15;
#pragma unroll
  for (int t = 0; t < NT; ++t) {
    const float bn = bias[n0 + t * 16 + l15];
#pragma unroll
    for (int r = 0; r < 8; ++r)
      orow[(size_t)r * 128 + n0 + t * 16] = acc[t][r] + bn;
  }
}

// ---------------------------------------------------------------------------
// Brute-force 3-NN + inverse-distance interpolation, accumulated into `out`
// (which already holds the up-projection). Down points are staged through LDS
// in 2048-point tiles (32 KB as padded float4 -> single ds_load_b128 broadcast
// per candidate). One thread per query point keeps a register top-3.
// ---------------------------------------------------------------------------
#define KNN_TILE 2048

__global__ __launch_bounds__(256) void knn_interp_accum(
    const float* __restrict__ qpts,   // [M,3]
    const float* __restrict__ dpts,   // [N,3]
    const float* __restrict__ dfeat,  // [N,128]  (projected down features)
    float* __restrict__ out,          // [M,128]  (+= interpolation)
    int M, int N) {
  __shared__ float4 sp[KNN_TILE];

  const int m = blockIdx.x * blockDim.x + threadIdx.x;
  float qx = 0.f, qy = 0.f, qz = 0.f;
  if (m < M) {
    qx = qpts[3 * (size_t)m + 0];
    qy = qpts[3 * (size_t)m + 1];
    qz = qpts[3 * (size_t)m + 2];
  }

  float d0 = 3.4e38f, d1 = 3.4e38f, d2 = 3.4e38f;
  int   i0 = 0,       i1 = 0,       i2 = 0;

  for (int base = 0; base < N; base += KNN_TILE) {
    const int cnt = (N - base < KNN_TILE) ? (N - base) : KNN_TILE;
    __syncthreads();
    for (int t = threadIdx.x; t < cnt; t += blockDim.x) {
      float4 v;
      v.x = dpts[3 * (size_t)(base + t) + 0];
      v.y = dpts[3 * (size_t)(base + t) + 1];
      v.z = dpts[3 * (size_t)(base + t) + 2];
      v.w = 0.f;
      sp[t] = v;
    }
    __syncthreads();

#pragma unroll 4
    for (int j = 0; j < cnt; ++j) {
      float4 p = sp[j];                       // broadcast LDS read
      float dx = qx - p.x, dy = qy - p.y, dz = qz - p.z;
      float d  = fmaf(dx, dx, fmaf(dy, dy, dz * dz));
      int  gj  = base + j;
      if (d < d2) {                           // register top-3 insert
        if (d < d1) {
          d2 = d1; i2 = i1;
          if (d < d0) { d1 = d0; i1 = i0; d0 = d; i0 = gj; }
          else        { d1 = d;  i1 = gj; }
        } else      { d2 = d;  i2 = gj; }
      }
    }
  }

  if (m >= M) return;

  const float r0 = 1.0f / (d0 + 1e-8f);
  const float r1 = 1.0f / (d1 + 1e-8f);
  const float r2 = 1.0f / (d2 + 1e-8f);
  const float inv = 1.0f / (r0 + r1 + r2);
  const float w0 = r0 * inv, w1 = r1 * inv, w2 = r2 * inv;

  const float4* f0 = (const float4*)(dfeat + (size_t)i0 * 128);
  const float4* f1 = (const float4*)(dfeat + (size_t)i1 * 128);
  const float4* f2 = (const float4*)(dfeat + (size_t)i2 * 128);
  float4* o = (float4*)(out + (size_t)m * 128);
#pragma unroll 4
  for (int c = 0; c < 32; ++c) {
    float4 a = f0[c], b = f1[c], g = f2[c], r = o[c];
    r.x += w0 * a.x + w1 * b.x + w2 * g.x;
    r.y += w0 * a.y + w1 * b.y + w2 * g.y;
    r.z += w0 * a.z + w1 * b.z + w2 * g.z;
    r.w += w0 * a.w + w1 * b.w + w2 * g.w;
    o[c] = r;
  }
}

// ---------------------------------------------------------------------------
extern "C" void kernel_launch(void* const* d_in, const int* in_sizes, int n_in,
                              void* d_out, int out_size, void* d_ws,
                              size_t ws_size, hipStream_t stream) {
  const float* up_points     = (const float*)d_in[0];  // [M,3]
  const float* up_features   = (const float*)d_in[1];  // [M,128]
  const float* down_points   = (const float*)d_in[2];  // [N,3]
  const float* down_features = (const float*)d_in[3];  // [N,256]
  const float* W_up          = (const float*)d_in[4];  // [128,128]
  const float* b_up          = (const float*)d_in[5];  // [128]
  const float* W_down        = (const float*)d_in[6];  // [128,256]
  const float* b_down        = (const float*)d_in[7];  // [128]
  float* out = (float*)d_out;

  const int M = in_sizes[0] / 3;   // 32768
  const int N = in_sizes[2] / 3;   // 8192

  float* down_f = (float*)d_ws;    // [N,128] scratch, 4 MB

  // 1) down projection -> workspace. 16x64 slab per wave (NT=4), 4 waves/blk.
  {
    const int slabs  = (N / 16) * 2;           // 128 cols / 64 per slab
    const int blocks = (slabs + 3) / 4;
    gemm_bias_wmma_f32<256, 4><<<blocks, 128, 0, stream>>>(
        down_features, W_down, b_down, down_f, N);
  }

  // 2) up projection -> d_out
  {
    const int slabs  = (M / 16) * 2;
    const int blocks = (slabs + 3) / 4;
    gemm_bias_wmma_f32<128, 4><<<blocks, 128, 0, stream>>>(
        up_features, W_up, b_up, out, M);
  }

  // 3) 3-NN inverse-distance interpolation accumulated into d_out
  {
    const int blocks = (M + 255) / 256;
    knn_interp_accum<<<blocks, 256, 0, stream>>>(
        up_points, down_points, down_f, out, M, N);
  }
}